// EnsembleGNN_25409026524028
// MI455X (gfx1250) — compile-verified
//
#include <hip/hip_runtime.h>

#define N_MEMBERS 2
#define N_NODES   65536
#define N_EDGES   1048576
#define N_LAYERS  12
#define DDIM      64

typedef float v2f __attribute__((ext_vector_type(2)));
typedef float v8f __attribute__((ext_vector_type(8)));

// ---------------------------------------------------------------- utilities
__global__ void k_fill(float* __restrict__ p, float v, int n) {
    int i = blockIdx.x * blockDim.x + threadIdx.x;
    if (i < n) p[i] = v;
}

__global__ void k_deg_scatter(float* __restrict__ deg, const int* __restrict__ dst, int E) {
    int e = blockIdx.x * blockDim.x + threadIdx.x;
    if (e < E) atomicAdd(&deg[dst[e]], 1.0f);
}

__global__ void k_rsqrt_inplace(float* __restrict__ p, int n) {
    int i = blockIdx.x * blockDim.x + threadIdx.x;
    if (i < n) p[i] = rsqrtf(p[i]);
}

// ---------------------------------------------------------------- GEMM (WMMA)
// hw[N, 64] = relu?(h)[N, 64] @ W[64, 64]
// blockDim = 256 (8 wave32). Each wave computes a 16-row x 64-col strip using
// V_WMMA_F32_16X16X4_F32, iterating K in steps of 4 (16 steps x 4 N-tiles).
// W is staged in LDS pre-packed as K-pairs: Ws[k/2][n][2] = {W[k][n], W[k+1][n]},
// so each B fragment is a single aligned ds_load_b64 straight into a WMMA
// source register pair (no v_mov repacking).
template <bool RELU>
__global__ void __launch_bounds__(256)
k_gemm_wmma(const float* __restrict__ h, const float* __restrict__ Wg,
            float* __restrict__ hw) {
    __shared__ float Ws[32 * 128];              // 16 KB of the 320 KB WGP LDS
    for (int i = threadIdx.x; i < DDIM * DDIM; i += 256) {
        const int k = i >> 6;                   // row of W
        const int n = i & 63;                   // col of W
        Ws[(k >> 1) * 128 + n * 2 + (k & 1)] = Wg[i];
    }
    __syncthreads();

    const int wave = threadIdx.x >> 5;
    const int lane = threadIdx.x & 31;
    const int hi   = lane >> 4;                 // 0: K pair {k,k+1}; 1: {k+2,k+3}
    const int l    = lane & 15;
    const int m0   = (blockIdx.x * 8 + wave) * 16;

    v8f c0 = {}, c1 = {}, c2 = {}, c3 = {};
    const float* arow = h + (size_t)(m0 + l) * DDIM;

#pragma unroll
    for (int k = 0; k < DDIM; k += 4) {
        const int ka = k + hi * 2;              // always even
        // A fragment (ISA layout): lanes 0-15 hold K=k,k+1; lanes 16-31 K=k+2,k+3
        v2f a = *(const v2f*)(arow + ka);       // aligned global_load_b64
        if (RELU) { a.x = fmaxf(a.x, 0.0f); a.y = fmaxf(a.y, 0.0f); }

        const float* bp = Ws + (ka >> 1) * 128; // packed row of K-pairs
        v2f b0 = *(const v2f*)(bp + ( 0 + l) * 2);  // ds_load_b64, even pair
        v2f b1 = *(const v2f*)(bp + (16 + l) * 2);
        v2f b2 = *(const v2f*)(bp + (32 + l) * 2);
        v2f b3 = *(const v2f*)(bp + (48 + l) * 2);

        c0 = __builtin_amdgcn_wmma_f32_16x16x4_f32(false, a, false, b0, (short)0, c0, false, false);
        c1 = __builtin_amdgcn_wmma_f32_16x16x4_f32(false, a, false, b1, (short)0, c1, false, false);
        c2 = __builtin_amdgcn_wmma_f32_16x16x4_f32(false, a, false, b2, (short)0, c2, false, false);
        c3 = __builtin_amdgcn_wmma_f32_16x16x4_f32(false, a, false, b3, (short)0, c3, false, false);
    }

    // D layout: VGPR r -> lanes 0-15 row m0+r, lanes 16-31 row m0+8+r, col n0+l
    const int rowBase = m0 + hi * 8;
#pragma unroll
    for (int r = 0; r < 8; ++r) {
        float* orow = hw + (size_t)(rowBase + r) * DDIM + l;
        orow[ 0] = c0[r];
        orow[16] = c1[r];
        orow[32] = c2[r];
        orow[48] = c3[r];
    }
}

// ---------------------------------------------------------------- self-loop init
// agg[n, :] = dinv[n]^2 * hw[n, :] + b[:]     (fully overwrites agg)
__global__ void __launch_bounds__(256)
k_init_agg(float* __restrict__ agg, const float* __restrict__ hw,
           const float* __restrict__ dinv, const float* __restrict__ b) {
    int t  = blockIdx.x * blockDim.x + threadIdx.x;    // N_NODES*16 threads
    int n  = t >> 4;
    int c4 = (t & 15) * 4;
    float di = dinv[n];
    float s  = di * di;
    const float4 hv = *(const float4*)(hw + (size_t)n * DDIM + c4);
    const float4 bv = *(const float4*)(b + c4);
    float4 o;
    o.x = s * hv.x + bv.x;
    o.y = s * hv.y + bv.y;
    o.z = s * hv.z + bv.z;
    o.w = s * hv.w + bv.w;
    *(float4*)(agg + (size_t)n * DDIM + c4) = o;
}

// ---------------------------------------------------------------- edge scatter
// agg[dst, :] += dinv[src]*dinv[dst] * hw[src, :]
// 32 threads per edge (edge id uniform across the wave), 2 columns per thread.
__global__ void __launch_bounds__(256)
k_scatter(float* __restrict__ agg, const float* __restrict__ hw,
          const float* __restrict__ dinv, const int* __restrict__ src,
          const int* __restrict__ dst) {
    unsigned t = blockIdx.x * blockDim.x + threadIdx.x;  // N_EDGES*32 threads
    int e = (int)(t >> 5);
    int c = (int)(t & 31) * 2;
    int s = src[e];
    int d = dst[e];
    float w = dinv[s] * dinv[d];
    const float2 hv = *(const float2*)(hw + (size_t)s * DDIM + c);
    float* ap = agg + (size_t)d * DDIM + c;
    atomicAdd(ap,     w * hv.x);
    atomicAdd(ap + 1, w * hv.y);
}

// ---------------------------------------------------------------- launcher
extern "C" void kernel_launch(void* const* d_in, const int* in_sizes, int n_in,
                              void* d_out, int out_size, void* d_ws, size_t ws_size,
                              hipStream_t stream) {
    const float* x    = (const float*)d_in[0];   // [2, 65536, 64]
    const int*   edge = (const int*)d_in[1];     // [2, 2, 1048576] (int32 per harness)
    const float* W    = (const float*)d_in[2];   // [2, 12, 64, 64]
    const float* b    = (const float*)d_in[3];   // [2, 12, 64]
    float* out = (float*)d_out;                  // [2, 65536, 64]

    float* ws   = (float*)d_ws;
    float* dinv = ws;                                    //  64 K floats
    float* hw   = dinv + N_NODES;                        //   4 M floats
    float* h0   = hw + (size_t)N_NODES * DDIM;           //   4 M floats
    float* h1   = h0 + (size_t)N_NODES * DDIM;           //   4 M floats

    for (int i = 0; i < N_MEMBERS; ++i) {
        const int* src = edge + (size_t)i * 2 * N_EDGES;
        const int* dst = src + N_EDGES;

        // degree (with self-loop) -> dinv, once per member
        k_fill<<<N_NODES / 256, 256, 0, stream>>>(dinv, 1.0f, N_NODES);
        k_deg_scatter<<<N_EDGES / 256, 256, 0, stream>>>(dinv, dst, N_EDGES);
        k_rsqrt_inplace<<<N_NODES / 256, 256, 0, stream>>>(dinv, N_NODES);

        const float* h = x + (size_t)i * N_NODES * DDIM;
        for (int j = 0; j < N_LAYERS; ++j) {
            const float* Wj = W + ((size_t)i * N_LAYERS + j) * DDIM * DDIM;
            const float* bj = b + ((size_t)i * N_LAYERS + j) * DDIM;
            float* agg = (j == N_LAYERS - 1) ? (out + (size_t)i * N_NODES * DDIM)
                                             : ((j & 1) ? h1 : h0);

            // hw = relu?(h) @ Wj   (ReLU of previous layer folded into A load)
            if (j > 0)
                k_gemm_wmma<true><<<N_NODES / 128, 256, 0, stream>>>(h, Wj, hw);
            else
                k_gemm_wmma<false><<<N_NODES / 128, 256, 0, stream>>>(h, Wj, hw);
            // agg = dinv^2 * hw + b
            k_init_agg<<<(N_NODES * 16) / 256, 256, 0, stream>>>(agg, hw, dinv, bj);
            // agg[dst] += dinv[src]*dinv[dst] * hw[src]
            k_scatter<<<(N_EDGES * 32) / 256, 256, 0, stream>>>(agg, hw, dinv, src, dst);

            h = agg;
        }
    }
}